// Criterion_26250840113347
// MI455X (gfx1250) — compile-verified
//
#include <hip/hip_runtime.h>

#define B_    8
#define N_    4096
#define NCLS  13
#define YTILE 1024   // y points staged in LDS per chunk (YTILE/16 chunks * 32 lanes * 8B = 16 KB)

typedef __attribute__((ext_vector_type(2))) float v2f;
typedef __attribute__((ext_vector_type(8))) float v8f;

// ws[0] = sum of picked log-probs (seg), ws[1] = sum dist1 mins, ws[2] = sum dist2 mins
__global__ void init_ws_kernel(float* ws) {
    if (threadIdx.x < 4) ws[threadIdx.x] = 0.0f;
}

// NLL gather: sum seg_prob[b,n,label[b,n]] into ws[0]
__global__ __launch_bounds__(256) void seg_nll_kernel(const float* __restrict__ seg_prob,
                                                      const int* __restrict__ seg_label,
                                                      float* __restrict__ acc) {
    int idx = blockIdx.x * blockDim.x + threadIdx.x;   // 0 .. B_*N_-1
    float v = 0.0f;
    if (idx < B_ * N_) {
        int l = seg_label[idx];
        v = seg_prob[(size_t)idx * NCLS + l];
    }
    // wave32 reduction
    for (int m = 16; m > 0; m >>= 1) v += __shfl_xor(v, m, 32);
    if ((threadIdx.x & 31) == 0) atomicAdd(acc, v);
}

// For each row i of X (B,N,3): min over j of ||x_i - y_j||^2, summed into *acc.
// pair = ||x||^2 + A*B where A = [x0,x1,x2,1] (16x4), B = [-2y0,-2y1,-2y2,||y||^2] (4x16),
// computed with V_WMMA_F32_16X16X4_F32. One wave per 16-row strip, 8 waves/block.
// LDS holds B-operands pre-swizzled into WMMA lane layout: for chunk c (16 columns),
// slot [c*32 + lane]: lane<16 -> {-2y0,-2y1} of column lane; lane>=16 -> {-2y2,||y||^2}.
__global__ __launch_bounds__(256) void chamfer_min_kernel(const float* __restrict__ X,
                                                          const float* __restrict__ Y,
                                                          float* __restrict__ acc) {
    __shared__ v2f ytile[(YTILE / 16) * 32];

    const int b    = blockIdx.y;
    const int tid  = threadIdx.x;
    const int wave = tid >> 5;
    const int lane = tid & 31;
    const int half = lane >> 4;      // 0: lanes 0-15 (K=0,1 / rows M=0-7), 1: lanes 16-31
    const int l16  = lane & 15;

    const float* xb = X + (size_t)b * N_ * 3;
    const float* yb = Y + (size_t)b * N_ * 3;

    // A operand: row m = rowBase + l16 of X, padded with 1.0 in K=3
    const int rowBase = blockIdx.x * 128 + wave * 16;
    const int row     = rowBase + l16;
    const float px = xb[row * 3 + 0];
    const float py = xb[row * 3 + 1];
    const float pz = xb[row * 3 + 2];
    v2f a;
    a.x = half ? pz : px;
    a.y = half ? 1.0f : py;

    v8f mn;
#pragma unroll
    for (int k = 0; k < 8; ++k) mn[k] = 3.0e38f;

    for (int jbase = 0; jbase < N_; jbase += YTILE) {
        __syncthreads();
        // Cooperative stage into WMMA B-operand layout
        for (int j = tid; j < YTILE; j += 256) {
            const float* yp = yb + (size_t)(jbase + j) * 3;
            float y0 = yp[0], y1 = yp[1], y2 = yp[2];
            int c = j >> 4, l = j & 15;
            v2f lo; lo.x = -2.0f * y0; lo.y = -2.0f * y1;
            v2f hi; hi.x = -2.0f * y2; hi.y = y0 * y0 + y1 * y1 + y2 * y2;
            ytile[c * 32 + l]      = lo;
            ytile[c * 32 + 16 + l] = hi;
        }
        __syncthreads();

        // Sweep YTILE columns, 16 per WMMA; B operand is a direct 8-byte LDS load
#pragma unroll 4
        for (int c = 0; c < YTILE / 16; ++c) {
            v2f bb = ytile[c * 32 + lane];
            v8f cacc = {};
            cacc = __builtin_amdgcn_wmma_f32_16x16x4_f32(
                    /*neg_a=*/false, a, /*neg_b=*/false, bb,
                    /*c_mod=*/(short)0, cacc, /*reuse_a=*/false, /*reuse_b=*/false);
#pragma unroll
            for (int k = 0; k < 8; ++k) mn[k] = fminf(mn[k], cacc[k]);
        }
    }

    // Min across the 16 lanes of each half (xor masks < 16 stay within the half)
#pragma unroll
    for (int m = 8; m >= 1; m >>= 1) {
#pragma unroll
        for (int k = 0; k < 8; ++k) {
            float o = __shfl_xor(mn[k], m, 32);
            mn[k] = fminf(mn[k], o);
        }
    }

    // lane 0 holds row mins for M=0-7, lane 16 for M=8-15; add ||x||^2 and accumulate
    if (l16 == 0) {
        float s = 0.0f;
#pragma unroll
        for (int m = 0; m < 8; ++m) {
            int r = rowBase + half * 8 + m;
            float x0 = xb[r * 3 + 0], x1 = xb[r * 3 + 1], x2 = xb[r * 3 + 2];
            s += mn[m] + x0 * x0 + x1 * x1 + x2 * x2;
        }
        atomicAdd(acc, s);
    }
}

__global__ void finalize_kernel(const float* __restrict__ ws, float* __restrict__ out) {
    // seg_loss = -ws[0]/(B*N); cha = (ws[1]+ws[2])/(B*N)
    out[0] = (ws[1] + ws[2] - ws[0]) * (1.0f / (float)(B_ * N_));
}

extern "C" void kernel_launch(void* const* d_in, const int* in_sizes, int n_in,
                              void* d_out, int out_size, void* d_ws, size_t ws_size,
                              hipStream_t stream) {
    const float* seg_prob  = (const float*)d_in[0];
    const int*   seg_label = (const int*)d_in[1];
    const float* point_rec = (const float*)d_in[2];
    const float* point     = (const float*)d_in[3];
    float* ws  = (float*)d_ws;
    float* out = (float*)d_out;

    init_ws_kernel<<<1, 32, 0, stream>>>(ws);

    seg_nll_kernel<<<(B_ * N_) / 256, 256, 0, stream>>>(seg_prob, seg_label, ws + 0);

    dim3 grid(N_ / 128, B_);
    // dist1: rows = point_rec, cols = point
    chamfer_min_kernel<<<grid, 256, 0, stream>>>(point_rec, point, ws + 1);
    // dist2: rows = point, cols = point_rec (symmetric)
    chamfer_min_kernel<<<grid, 256, 0, stream>>>(point, point_rec, ws + 2);

    finalize_kernel<<<1, 1, 0, stream>>>(ws, out);
}